// Sampler_57140244906458
// MI455X (gfx1250) — compile-verified
//
#include <hip/hip_runtime.h>
#include <hip/hip_bf16.h>

#define B        256
#define V        128000
#define THREADS  512
#define TILE     2048           // floats per staged tile (8 KB); double-buffered
#define NBINS    2048
#define NTOP     5

typedef __attribute__((ext_vector_type(4))) unsigned int u32x4;
typedef __attribute__((ext_vector_type(8))) int          i32x8;
typedef __attribute__((ext_vector_type(4))) int          i32x4;

#if defined(__HIP_DEVICE_COMPILE__) && defined(__has_builtin)
#if __has_builtin(__builtin_amdgcn_tensor_load_to_lds) && __has_builtin(__builtin_amdgcn_s_wait_tensorcnt)
#define HAVE_TDM 1
#endif
#endif
#ifndef HAVE_TDM
#define HAVE_TDM 0
#endif

// order-preserving float -> uint key (larger float => larger key)
__device__ __forceinline__ unsigned f2k(float f) {
  unsigned u = __float_as_uint(f);
  return (u & 0x80000000u) ? ~u : (u | 0x80000000u);
}

// deterministic Gumbel(0,1) noise from a 64-bit hash (splitmix64)
__device__ __forceinline__ float gumbel_noise(unsigned long long h) {
  h += 0x9E3779B97F4A7C15ull;
  h = (h ^ (h >> 30)) * 0xBF58476D1CE4E5B9ull;
  h = (h ^ (h >> 27)) * 0x94D049BB133111EBull;
  h ^= h >> 31;
  float u = ((float)(unsigned)(h >> 40) + 0.5f) * (1.0f / 16777216.0f); // (0,1)
  return -__logf(-__logf(u));
}

#if HAVE_TDM
// Issue one TDM 1-D tile load: `count` floats from gsrc -> LDS byte offset.
// D# per CDNA5 ISA Sec 8.3/8.4 (group0: count=1,type=2; group1: 4B elems,
// tensor_dim0 = tile_dim0 = count, tile_dim1 = 1).
__device__ __forceinline__ void tdm_stage_1d(const float* gsrc, unsigned lds_byte, int count) {
  unsigned long long ga = (unsigned long long)(size_t)gsrc;
  unsigned n = (unsigned)count;
  u32x4 g0;
  g0[0] = 1u;                                                  // valid user descriptor
  g0[1] = lds_byte;                                            // lds_addr (bytes)
  g0[2] = (unsigned)ga;                                        // global_addr[31:0]
  g0[3] = (unsigned)((ga >> 32) & 0x1FFFFFFull) | (2u << 30);  // addr[56:32] | type=2
  i32x8 g1;
  g1[0] = (int)(2u << 16);                    // data_size = 4 bytes, wg_mask = 0
  g1[1] = (int)((n & 0xFFFFu) << 16);         // tensor_dim0[15:0]
  g1[2] = (int)((n >> 16) | (1u << 16));      // tensor_dim0[31:16] | tensor_dim1 = 1
  g1[3] = (int)(n << 16);                     // tile_dim0 = n
  g1[4] = 1;                                  // tile_dim1 = 1, tile_dim2 = 0
  g1[5] = (int)n;                             // tensor_dim0_stride (lo)
  g1[6] = 0;
  g1[7] = 0;
  i32x4 z4 = {0, 0, 0, 0};
#if __clang_major__ >= 23
  i32x8 z8 = {0, 0, 0, 0, 0, 0, 0, 0};
  __builtin_amdgcn_tensor_load_to_lds(g0, g1, z4, z4, z8, 0);
#else
  __builtin_amdgcn_tensor_load_to_lds(g0, g1, z4, z4, 0);
#endif
}
#endif

// Stream one row through a double-buffered LDS tile ring staged by the TDM
// (wave 0 issues; TDM ignores EXEC). fn(value, vocab_index) is called for
// every element. Tensor ops complete in-order, so we issue tile t+1 BEFORE
// waiting tensorcnt<=1 for tile t: the wait returns while t+1 is in flight.
// Tiles are consumed as float4 (ds_load_b128): one b128 per thread per tile.
template <typename F>
__device__ void stream_row(const float* __restrict__ row, float* tilebuf, F&& fn) {
#if HAVE_TDM
  const int nt = (V + TILE - 1) / TILE;
  const bool issuer = (threadIdx.x >> 5) == 0; // wave 0 only
  const unsigned lds0 = (unsigned)(size_t)(void*)tilebuf; // low 32 bits = LDS offset
  if (issuer) tdm_stage_1d(row, lds0, TILE < V ? TILE : V);
  for (int t = 0; t < nt; ++t) {
    if (issuer) {
      if (t + 1 < nt) {
        int base = (t + 1) * TILE;
        int cnt = V - base; if (cnt > TILE) cnt = TILE;
        tdm_stage_1d(row + base, lds0 + (unsigned)(((t + 1) & 1) * TILE * 4), cnt);
        __builtin_amdgcn_s_wait_tensorcnt(1);  // tile t done; t+1 still in flight
      } else {
        __builtin_amdgcn_s_wait_tensorcnt(0);
      }
    }
    __syncthreads(); // tile t visible to all waves
    const float4* buf4 = (const float4*)(tilebuf + (t & 1) * TILE);
    const int base = t * TILE;
    int cnt = V - base; if (cnt > TILE) cnt = TILE;
    const int nq = cnt >> 2;                   // cnt is a multiple of 4 here
    for (int q = (int)threadIdx.x; q < nq; q += THREADS) {
      float4 w = buf4[q];
      int gi = base + q * 4;
      fn(w.x, gi); fn(w.y, gi + 1); fn(w.z, gi + 2); fn(w.w, gi + 3);
    }
    for (int i = base + (nq << 2) + (int)threadIdx.x; i < base + cnt; i += THREADS)
      fn(tilebuf[(t & 1) * TILE + (i - base)], i);  // safety tail (normally empty)
    __syncthreads();
  }
#else
  for (int i = (int)threadIdx.x; i < V; i += THREADS) fn(row[i], i);
  __syncthreads();
#endif
}

__global__ __launch_bounds__(THREADS) void sampler_kernel(
    const float* __restrict__ logits, const float* __restrict__ temperature,
    const int* __restrict__ topk, const float* __restrict__ topp,
    int* __restrict__ out_sampled, float* __restrict__ out_lp,
    int* __restrict__ out_idx) {
  __shared__ __align__(16) float s_tiles[2 * TILE];
  __shared__ unsigned s_hist[NBINS];
  __shared__ float    s_f1[THREADS];
  __shared__ float    s_f2[THREADS];
  __shared__ int      s_i1[THREADS];
  __shared__ float    s_p5v[THREADS * NTOP];
  __shared__ int      s_p5i[THREADS * NTOP];
  __shared__ unsigned s_prefix, s_krem, s_Tk, s_bp1, s_T;
  __shared__ float    s_m, s_tau, s_Zf, s_bx;
  __shared__ int      s_bi;
  __shared__ float    s_wv[NTOP];
  __shared__ int      s_wi[NTOP];

  float* histF = (float*)s_hist;
  const int row = blockIdx.x;
  const int tid = (int)threadIdx.x;
  const float NEG_INF = __int_as_float(0xff800000u);

  const float invT = 1.0f / temperature[row];
  const int   kwant = topk[row];
  const float p = topp[row];
  const float* __restrict__ rowp = logits + (size_t)row * V;

  // ---------- Pass A: row max + radix round 1 (key bits [31:21]) ----------
  for (int i = tid; i < NBINS; i += THREADS) s_hist[i] = 0u;
  float lmax = NEG_INF;
  stream_row(rowp, s_tiles, [&](float v, int idx) {
    (void)idx;
    float x = v * invT;
    if (x > lmax) lmax = x;
    atomicAdd(&s_hist[f2k(x) >> 21], 1u);
  });
  s_f1[tid] = lmax;
  __syncthreads();
  for (int s = THREADS >> 1; s > 0; s >>= 1) {
    if (tid < s && s_f1[tid + s] > s_f1[tid]) s_f1[tid] = s_f1[tid + s];
    __syncthreads();
  }
  if (tid == 0) {
    s_m = s_f1[0];
    unsigned rem = (unsigned)kwant, cum = 0, sel = 0;
    for (int i = NBINS - 1; i >= 0; --i) {
      unsigned c = s_hist[i];
      if (cum + c >= rem) { sel = (unsigned)i; rem -= cum; break; }
      cum += c;
    }
    s_prefix = sel; s_krem = rem;
  }
  __syncthreads();

  // ---------- Pass B: radix round 2 (key bits [20:10]) ----------
  const unsigned pfx1 = s_prefix;
  for (int i = tid; i < NBINS; i += THREADS) s_hist[i] = 0u;
  stream_row(rowp, s_tiles, [&](float v, int idx) {
    (void)idx;
    unsigned key = f2k(v * invT);
    if ((key >> 21) == pfx1) atomicAdd(&s_hist[(key >> 10) & 0x7FFu], 1u);
  });
  if (tid == 0) {
    unsigned rem = s_krem, cum = 0, sel = 0;
    for (int i = NBINS - 1; i >= 0; --i) {
      unsigned c = s_hist[i];
      if (cum + c >= rem) { sel = (unsigned)i; rem -= cum; break; }
      cum += c;
    }
    s_prefix = (pfx1 << 11) | sel; s_krem = rem;
  }
  __syncthreads();

  // ---------- Pass C: radix round 3 (key bits [9:0]) -> exact k-th key ----------
  const unsigned pfx2 = s_prefix;
  for (int i = tid; i < NBINS; i += THREADS) s_hist[i] = 0u;
  stream_row(rowp, s_tiles, [&](float v, int idx) {
    (void)idx;
    unsigned key = f2k(v * invT);
    if ((key >> 10) == pfx2) atomicAdd(&s_hist[key & 0x3FFu], 1u);
  });
  if (tid == 0) {
    unsigned rem = s_krem, cum = 0, sel = 0;
    for (int i = 1023; i >= 0; --i) {
      unsigned c = s_hist[i];
      if (cum + c >= rem) { sel = (unsigned)i; break; }
      cum += c;
    }
    s_Tk = (pfx2 << 10) | sel;  // keep: key >= Tk  (ties kept, matches reference)
  }
  __syncthreads();

  const unsigned Tk = s_Tk;
  const float    m  = s_m;

  // ---------- Pass E: Z over top-k survivors + weighted quantile round 1 ----------
  for (int i = tid; i < NBINS; i += THREADS) histF[i] = 0.0f;
  stream_row(rowp, s_tiles, [&](float v, int idx) {
    (void)idx;
    float x = v * invT;
    unsigned key = f2k(x);
    if (key >= Tk) atomicAdd(&histF[key >> 21], __expf(x - m));
  });
  if (tid == 0) {
    float Z = 0.0f;
    for (int i = 0; i < NBINS; ++i) Z += histF[i];
    float tau = (1.0f - p) * Z;          // tail mass to drop (ascending)
    float cum = 0.0f, t2 = 0.0f;
    unsigned sel = 0;
    for (int i = 0; i < NBINS; ++i) {
      float c = histF[i];
      if (cum + c > tau) { sel = (unsigned)i; t2 = tau - cum; break; }
      cum += c;
    }
    s_bp1 = sel; s_tau = t2;
  }
  __syncthreads();

  // ---------- Pass F: weighted quantile round 2 -> top-p threshold ----------
  const unsigned bp1 = s_bp1;
  const float    tau2 = s_tau;
  for (int i = tid; i < NBINS; i += THREADS) histF[i] = 0.0f;
  stream_row(rowp, s_tiles, [&](float v, int idx) {
    (void)idx;
    float x = v * invT;
    unsigned key = f2k(x);
    if (key >= Tk && (key >> 21) == bp1)
      atomicAdd(&histF[(key >> 10) & 0x7FFu], __expf(x - m));
  });
  if (tid == 0) {
    float cum = 0.0f;
    unsigned sel = 0x7FFu;
    for (int i = 0; i < NBINS; ++i) {
      float c = histF[i];
      if (cum + c > tau2) { sel = (unsigned)i; break; }
      cum += c;
    }
    unsigned Tp = ((bp1 << 11) | sel) << 10;  // lower edge of boundary sub-bin
    s_T = (Tp > Tk) ? Tp : Tk;
  }
  __syncthreads();

  // ---------- Pass G: Zf + Gumbel-max sample + per-thread top-5 ----------
  const unsigned T = s_T;
  float zpart = 0.0f;
  float t5v[NTOP]; int t5i[NTOP];
  for (int j = 0; j < NTOP; ++j) { t5v[j] = NEG_INF; t5i[j] = -1; }
  float bscore = NEG_INF, bx = NEG_INF; int bi = 0;
  const unsigned long long seed =
      ((unsigned long long)row << 32) ^ 0xC0FFEE1234ABCDull;
  stream_row(rowp, s_tiles, [&](float v, int idx) {
    float x = v * invT;
    unsigned key = f2k(x);
    if (key >= T) {
      zpart += __expf(x - m);
      float sc = x + gumbel_noise(seed + (unsigned long long)(unsigned)idx);
      if (sc > bscore) { bscore = sc; bx = x; bi = idx; }
      if (x > t5v[NTOP - 1]) {
        int j = NTOP - 1;
        while (j > 0 && t5v[j - 1] < x) {
          t5v[j] = t5v[j - 1]; t5i[j] = t5i[j - 1]; --j;
        }
        t5v[j] = x; t5i[j] = idx;
      }
    }
  });

  // Zf reduction
  s_f1[tid] = zpart;
  __syncthreads();
  for (int s = THREADS >> 1; s > 0; s >>= 1) {
    if (tid < s) s_f1[tid] += s_f1[tid + s];
    __syncthreads();
  }
  if (tid == 0) s_Zf = s_f1[0];
  __syncthreads();

  // Gumbel argmax reduction (score with {x, idx} payload)
  s_f1[tid] = bscore; s_f2[tid] = bx; s_i1[tid] = bi;
  __syncthreads();
  for (int s = THREADS >> 1; s > 0; s >>= 1) {
    if (tid < s && s_f1[tid + s] > s_f1[tid]) {
      s_f1[tid] = s_f1[tid + s]; s_f2[tid] = s_f2[tid + s]; s_i1[tid] = s_i1[tid + s];
    }
    __syncthreads();
  }
  if (tid == 0) { s_bx = s_f2[0]; s_bi = s_i1[0]; }
  __syncthreads();

  // Top-5 merge: pool of per-thread top-5, extract global max 5 times
  for (int j = 0; j < NTOP; ++j) {
    s_p5v[tid * NTOP + j] = t5v[j];
    s_p5i[tid * NTOP + j] = t5i[j];
  }
  __syncthreads();
  for (int r = 0; r < NTOP; ++r) {
    float mv = NEG_INF; int mp = tid * NTOP;
    for (int j = 0; j < NTOP; ++j) {
      float c = s_p5v[tid * NTOP + j];
      if (c > mv) { mv = c; mp = tid * NTOP + j; }
    }
    s_f1[tid] = mv; s_i1[tid] = mp;
    __syncthreads();
    for (int s = THREADS >> 1; s > 0; s >>= 1) {
      if (tid < s && s_f1[tid + s] > s_f1[tid]) {
        s_f1[tid] = s_f1[tid + s]; s_i1[tid] = s_i1[tid + s];
      }
      __syncthreads();
    }
    if (tid == 0) {
      int pos = s_i1[0];
      s_wv[r] = s_f1[0];
      s_wi[r] = s_p5i[pos];
      s_p5v[pos] = NEG_INF;  // remove winner from pool
    }
    __syncthreads();
  }

  if (tid == 0) {
    float lz = __logf(s_Zf);
    out_sampled[row] = s_bi;
    for (int j = 0; j < NTOP; ++j) {
      out_lp[row * (NTOP + 1) + j]  = s_wv[j] - m - lz;
      out_idx[row * (NTOP + 1) + j] = s_wi[j];
    }
    out_lp[row * (NTOP + 1) + NTOP]  = s_bx - m - lz;
    out_idx[row * (NTOP + 1) + NTOP] = s_bi;
  }
}

extern "C" void kernel_launch(void* const* d_in, const int* in_sizes, int n_in,
                              void* d_out, int out_size, void* d_ws, size_t ws_size,
                              hipStream_t stream) {
  (void)in_sizes; (void)n_in; (void)out_size; (void)d_ws; (void)ws_size;
  const float* logits      = (const float*)d_in[0];
  const float* temperature = (const float*)d_in[1];
  const int*   k           = (const int*)d_in[2];
  const float* p           = (const float*)d_in[3];
  // d_out layout (flat, return order): sampled[B] (int bits), lp[B*6] (float),
  // idx[B*6] (int bits)
  int*   out_sampled = (int*)d_out;
  float* out_lp      = (float*)d_out + B;
  int*   out_idx     = (int*)d_out + B + B * (NTOP + 1);
  hipLaunchKernelGGL(sampler_kernel, dim3(B), dim3(THREADS), 0, stream,
                     logits, temperature, k, p, out_sampled, out_lp, out_idx);
}